// SharpenUSMFilter_30150670417963
// MI455X (gfx1250) — compile-verified
//
#include <hip/hip_runtime.h>

typedef __attribute__((ext_vector_type(2))) float v2f;
typedef __attribute__((ext_vector_type(8))) float v8f;
typedef __attribute__((ext_vector_type(4))) unsigned int u32x4;
typedef __attribute__((ext_vector_type(4))) int i32x4;
typedef __attribute__((ext_vector_type(8))) int i32x8;

constexpr int IMG_H = 512;
constexpr int IMG_W = 512;
constexpr int TILE  = 64;            // output tile per block
constexpr int HALO_DIM = TILE + 4;   // 68
constexpr int HSTRIDE  = 68;         // TDM writes contiguous rows; 8*68 % 64 == 32 -> finalize conflict-free
constexpr int TSTRIDE  = 80;         // 2*80 % 64 == 32 -> B-fragment lane-halves hit disjoint banks

#if defined(__has_builtin)
# if __has_builtin(__builtin_amdgcn_tensor_load_to_lds)
#  define HAVE_TDM 1
# endif
# if __has_builtin(__builtin_amdgcn_s_wait_tensorcnt)
#  define WAIT_TENSORCNT0() __builtin_amdgcn_s_wait_tensorcnt(0)
# endif
#endif
#ifndef WAIT_TENSORCNT0
# define WAIT_TENSORCNT0() asm volatile("s_wait_tensorcnt 0x0" ::: "memory")
#endif

// jnp.pad mode='reflect': -1->1, -2->2, n->n-2, n+1->n-3
__device__ __forceinline__ int reflect_idx(int x, int n) {
    if (x < 0)  x = -x;
    if (x >= n) x = 2 * n - 2 - x;
    return x;
}

// kk[d] for d in [0,4] else 0 (taps symmetric: {k0,k1,k2,k1,k0}), branchless
__device__ __forceinline__ float kband(int d, float k0, float k1, float k2) {
    float r = 0.0f;
    r = (d == 0 || d == 4) ? k0 : r;
    r = (d == 1 || d == 3) ? k1 : r;
    r = (d == 2)           ? k2 : r;
    return r;
}

// Generic (border) halo loader: per-lane loads with reflect indexing.
__device__ __forceinline__ void load_halo_reflect(float* halo, const float* src,
                                                  int i0, int j0, int tid) {
    for (int idx = tid; idx < HALO_DIM * HALO_DIM; idx += 256) {
        int r  = idx / HALO_DIM;
        int c  = idx - r * HALO_DIM;
        int gi = reflect_idx(i0 - 2 + r, IMG_H);
        int gj = reflect_idx(j0 - 2 + c, IMG_W);
        halo[r * HSTRIDE + c] = src[gi * IMG_W + gj];
    }
}

__global__ __launch_bounds__(256)
void usm_sharpen_kernel(const float* __restrict__ img,
                        const float* __restrict__ params,
                        float* __restrict__ out) {
    __shared__ float halo[HALO_DIM * HSTRIDE];  // 68 x 68, contiguous rows (TDM layout)
    __shared__ float tmp [HALO_DIM * TSTRIDE];  // 68 x 64 horizontally blurred, stride 80

    const int tid  = threadIdx.x;
    const int bc   = blockIdx.y;              // b*3 + c
    const int b    = bc / 3;
    const int tile = blockIdx.x;              // 0..63
    const int i0   = (tile >> 3) * TILE;
    const int j0   = (tile & 7)  * TILE;

    const float* __restrict__ src = img + (size_t)bc * IMG_H * IMG_W;
    float*       __restrict__ dst = out + (size_t)bc * IMG_H * IMG_W;

    // Per-batch Gaussian taps: sigma = 2*p0 (clamped), amount = 2*p1
    float sigma  = fmaxf(2.0f * params[b * 2 + 0], 0.001f);
    float amount = 2.0f * params[b * 2 + 1];
    float inv2s2 = 1.0f / (2.0f * sigma * sigma);
    float e1 = expf(-1.0f * inv2s2);
    float e2 = expf(-4.0f * inv2s2);
    float norm = 1.0f / (1.0f + 2.0f * e1 + 2.0f * e2);
    float k0 = e2 * norm, k1 = e1 * norm, k2 = norm;

    // ---- Phase 1: stage 68x68 halo tile into LDS ----
    const bool interior = (i0 >= 2) && (i0 + TILE + 2 <= IMG_H) &&
                          (j0 >= 2) && (j0 + TILE + 2 <= IMG_W);
#ifdef HAVE_TDM
    if (interior) {
        if (tid < 32) {   // one wave issues a single 2-D TDM tile load
            const float* gsrc = src + (size_t)(i0 - 2) * IMG_W + (j0 - 2);
            unsigned long long ga = (unsigned long long)(uintptr_t)gsrc;
            unsigned int lds_base = (unsigned int)(uintptr_t)(&halo[0]);

            // D# group 0: count=1 | lds_addr | global_addr[56:0] | type=2
            u32x4 g0;
            g0[0] = 1u;
            g0[1] = lds_base;
            g0[2] = (unsigned int)(ga & 0xFFFFFFFFu);
            g0[3] = ((unsigned int)(ga >> 32) & 0x01FFFFFFu) | (2u << 30);

            // D# group 1: data_size=4B, tensor 512x512, tile 68x68, dim0 stride 512
            i32x8 g1;
            g1[0] = (int)(2u << 16);                               // data_size=2 (4 bytes)
            g1[1] = (int)((IMG_W & 0xFFFF) << 16);                 // tensor_dim0[15:0]
            g1[2] = (int)(((unsigned)IMG_W >> 16) | ((IMG_H & 0xFFFF) << 16)); // dim0 hi | dim1 lo
            g1[3] = (int)(((unsigned)IMG_H >> 16) | ((unsigned)HALO_DIM << 16)); // dim1 hi | tile_dim0
            g1[4] = (int)HALO_DIM;                                 // tile_dim1 (tile_dim2 = 0)
            g1[5] = (int)IMG_W;                                    // tensor_dim0_stride[31:0]
            g1[6] = 0;                                             // stride hi | dim1_stride lo
            g1[7] = 0;

            i32x4 gz = {0, 0, 0, 0};
#if __clang_major__ >= 23
            i32x8 gz8 = {0, 0, 0, 0, 0, 0, 0, 0};
            __builtin_amdgcn_tensor_load_to_lds(g0, g1, gz, gz, gz8, 0);
#else
            __builtin_amdgcn_tensor_load_to_lds(g0, g1, gz, gz, 0);
#endif
            WAIT_TENSORCNT0();
        }
    } else {
        load_halo_reflect(halo, src, i0, j0, tid);
    }
#else
    (void)interior;
    load_halo_reflect(halo, src, i0, j0, tid);
#endif
    __syncthreads();

    // ---- Phase 2: horizontal 5-tap pass -> tmp (68 rows x 64 cols) ----
    // 4 row-groups x 64 cols; each thread walks 17 rows with incremental offsets.
    {
        const int rg = tid >> 6;          // 0..3
        const int jc = tid & 63;
        int hoff = (rg * 17) * HSTRIDE + jc;
        int toff = (rg * 17) * TSTRIDE + jc;
        for (int i = 0; i < 17; ++i) {
            const float* h = &halo[hoff];
            tmp[toff] = k0 * (h[0] + h[4]) + k1 * (h[1] + h[3]) + k2 * h[2];
            hoff += HSTRIDE;
            toff += TSTRIDE;
        }
    }
    __syncthreads();

    // ---- Phase 3: vertical pass as banded matmul via V_WMMA_F32_16X16X4_F32 ----
    // blur(16x16) = N(16x20) * tmp_sub(20x16), N[m][K] = kk[K-m]
    const int lane = tid & 31;
    const int wave = tid >> 5;
    const int n    = lane & 15;
    const int hi   = lane >> 4;

    v2f afrag[5];
#pragma unroll
    for (int t = 0; t < 5; ++t) {
        int kk0 = 4 * t + 2 * hi;         // A: vgpr j, half hi -> K = 4t + 2*hi + j
        afrag[t][0] = kband(kk0     - n, k0, k1, k2);
        afrag[t][1] = kband(kk0 + 1 - n, k0, k1, k2);
    }

#pragma unroll
    for (int sidx = 0; sidx < 2; ++sidx) {
        int s  = wave + sidx * 8;          // wave handles subtiles s and s+8 of 4x4 grid
        int rb = (s >> 2) * 16;
        int cb = (s & 3)  * 16;

        v8f acc = {};
#pragma unroll
        for (int t = 0; t < 5; ++t) {
            int rowbase = rb + 4 * t + 2 * hi;   // B: vgpr j, half hi -> row K = 4t + 2*hi + j
            v2f bfrag;
            bfrag[0] = tmp[(rowbase + 0) * TSTRIDE + cb + n];
            bfrag[1] = tmp[(rowbase + 1) * TSTRIDE + cb + n];
            acc = __builtin_amdgcn_wmma_f32_16x16x4_f32(
                false, afrag[t], false, bfrag, (short)0, acc, false, false);
        }

        // ---- Phase 4: out = clip(orig + amount*(orig - blur), 0, 1) ----
#pragma unroll
        for (int v = 0; v < 8; ++v) {
            int m  = v + 8 * hi;
            int il = rb + m;
            int jl = cb + n;
            float orig = halo[(il + 2) * HSTRIDE + (jl + 2)];
            float val  = orig + amount * (orig - acc[v]);
            val = fminf(fmaxf(val, 0.0f), 1.0f);
            dst[(i0 + il) * IMG_W + (j0 + jl)] = val;
        }
    }
}

extern "C" void kernel_launch(void* const* d_in, const int* in_sizes, int n_in,
                              void* d_out, int out_size, void* d_ws, size_t ws_size,
                              hipStream_t stream) {
    (void)in_sizes; (void)n_in; (void)out_size; (void)d_ws; (void)ws_size;
    const float* img    = (const float*)d_in[0];
    const float* params = (const float*)d_in[1];
    float* out          = (float*)d_out;

    dim3 grid(64, 48);   // 8x8 tiles of 64x64 over 512x512, times B*C = 48 planes
    usm_sharpen_kernel<<<grid, 256, 0, stream>>>(img, params, out);
}